// FAGCN_20375324852686
// MI455X (gfx1250) — compile-verified
//
#include <hip/hip_runtime.h>
#include <hip/hip_bf16.h>

// Problem constants (from reference)
#define N_NODES  65536
#define F_IN     512
#define H_DIM    64
#define C_OUT    40
#define E_EDGES  1048576
#define EPS_FA   0.1f

typedef float v2f __attribute__((ext_vector_type(2)));
typedef float v8f __attribute__((ext_vector_type(8)));

// ---------------------------------------------------------------- utilities
__global__ void k_zero(float* __restrict__ p, int n) {
    int i = blockIdx.x * blockDim.x + threadIdx.x;
    if (i < n) p[i] = 0.0f;
}

// deg[dst] += 1 per edge (float, matching reference segment_sum of ones)
__global__ void k_deg(const int* __restrict__ ei, float* __restrict__ deg) {
    int e = blockIdx.x * blockDim.x + threadIdx.x;
    if (e < E_EDGES) {
        int d = ei[E_EDGES + e];
        unsafeAtomicAdd(&deg[d], 1.0f);
    }
}

// dis = deg>0 ? rsqrt(max(deg,1)) : 0
__global__ void k_dis(const float* __restrict__ deg, float* __restrict__ dis) {
    int i = blockIdx.x * blockDim.x + threadIdx.x;
    if (i < N_NODES) {
        float d = deg[i];
        dis[i] = (d > 0.0f) ? rsqrtf(fmaxf(d, 1.0f)) : 0.0f;
    }
}

// ------------------------------------------------ GEMM1: h0 = relu(x@W1+b1)
// One wave computes a 16-row x 64-col output strip using 4 accumulators so
// each row of x (128 MB total, the dominant traffic) is read exactly once.
// W1 (128 KB) stays hot in L2. fp32 WMMA (16x16x4) keeps reference precision;
// the kernel is HBM-bound so the low-rate f32 matrix path is free.
__global__ __launch_bounds__(256) void k_gemm1(const float* __restrict__ x,
                                               const float* __restrict__ W1,
                                               const float* __restrict__ b1,
                                               float* __restrict__ h0) {
    const int lane   = threadIdx.x & 31;
    const int wave   = threadIdx.x >> 5;
    const int tile   = blockIdx.x * 8 + wave;   // 4096 row tiles
    const int rowB   = tile * 16;
    const int lhalf  = lane >> 4;               // 0 or 1
    const int l15    = lane & 15;

    v8f acc0 = {}, acc1 = {}, acc2 = {}, acc3 = {};

    // A layout (16x4 f32): VGPR0 -> K = k+2*lhalf, VGPR1 -> K = k+2*lhalf+1,
    // row M = lane&15.  B layout mirrored: VGPR r holds W1 row (k+2*lhalf+r),
    // column N = lane&15 (+ 16*t for each column tile t).
    const float* xrow = x + (size_t)(rowB + l15) * F_IN + 2 * lhalf;

    for (int k = 0; k < F_IN; k += 4) {
        v2f a;
        a.x = xrow[k];
        a.y = xrow[k + 1];

        const float* wp = W1 + (size_t)(k + 2 * lhalf) * H_DIM + l15;
        v2f b0, b1v, b2v, b3v;
        b0.x  = wp[0];           b0.y  = wp[H_DIM];
        b1v.x = wp[16];          b1v.y = wp[H_DIM + 16];
        b2v.x = wp[32];          b2v.y = wp[H_DIM + 32];
        b3v.x = wp[48];          b3v.y = wp[H_DIM + 48];

        acc0 = __builtin_amdgcn_wmma_f32_16x16x4_f32(false, a, false, b0,
                                                     (short)0, acc0, false, false);
        acc1 = __builtin_amdgcn_wmma_f32_16x16x4_f32(false, a, false, b1v,
                                                     (short)0, acc1, false, false);
        acc2 = __builtin_amdgcn_wmma_f32_16x16x4_f32(false, a, false, b2v,
                                                     (short)0, acc2, false, false);
        acc3 = __builtin_amdgcn_wmma_f32_16x16x4_f32(false, a, false, b3v,
                                                     (short)0, acc3, false, false);
    }

    // C/D layout: VGPR r -> M = r (lanes 0-15) or r+8 (lanes 16-31), N = lane&15
#pragma unroll
    for (int r = 0; r < 8; ++r) {
        int row = rowB + r + 8 * lhalf;
        float* hp = h0 + (size_t)row * H_DIM + l15;
        hp[0]  = fmaxf(acc0[r] + b1[l15],      0.0f);
        hp[16] = fmaxf(acc1[r] + b1[l15 + 16], 0.0f);
        hp[32] = fmaxf(acc2[r] + b1[l15 + 32], 0.0f);
        hp[48] = fmaxf(acc3[r] + b1[l15 + 48], 0.0f);
    }
}

// ------------------------------------------- per-node attention dot products
// wave per node; lane j covers features j and j+32; shfl_xor tree reduction
__global__ __launch_bounds__(256) void k_att(const float* __restrict__ h,
                                             const float* __restrict__ attl,
                                             const float* __restrict__ attr,
                                             float* __restrict__ al,
                                             float* __restrict__ ar) {
    int lane = threadIdx.x & 31;
    int n = (blockIdx.x * blockDim.x + threadIdx.x) >> 5;
    if (n >= N_NODES) return;
    float hv0 = h[(size_t)n * H_DIM + lane];
    float hv1 = h[(size_t)n * H_DIM + lane + 32];
    float sl = hv0 * attl[lane] + hv1 * attl[lane + 32];
    float sr = hv0 * attr[lane] + hv1 * attr[lane + 32];
#pragma unroll
    for (int off = 16; off > 0; off >>= 1) {
        sl += __shfl_xor(sl, off, 32);
        sr += __shfl_xor(sr, off, 32);
    }
    if (lane == 0) { al[n] = sl; ar[n] = sr; }
}

// hout = EPS * h0 (re-initialized every call -> replay-deterministic)
__global__ void k_scale_init(const float4* __restrict__ h0,
                             float4* __restrict__ hout, int n4) {
    int i = blockIdx.x * blockDim.x + threadIdx.x;
    if (i < n4) {
        float4 v = h0[i];
        v.x *= EPS_FA; v.y *= EPS_FA; v.z *= EPS_FA; v.w *= EPS_FA;
        hout[i] = v;
    }
}

// ------------------------------------------- edge gather + atomic scatter-add
// One wave per edge; lane j handles features j and j+32. This kernel is the
// roofline of the whole workload: ~1 GB of gather+scatter traffic across both
// layers (~45 us at 23.3 TB/s).
__global__ __launch_bounds__(256) void k_edge(const int* __restrict__ ei,
                                              const float* __restrict__ dis,
                                              const float* __restrict__ al,
                                              const float* __restrict__ ar,
                                              const float* __restrict__ hin,
                                              float* __restrict__ hout) {
    int lane = threadIdx.x & 31;
    int e = (blockIdx.x * blockDim.x + threadIdx.x) >> 5;
    if (e >= E_EDGES) return;
    int s = ei[e];
    int d = ei[E_EDGES + e];
    float w = tanhf(al[s] + ar[d]) * dis[s] * dis[d];
    const float* hs = hin + (size_t)s * H_DIM;
    float* hd = hout + (size_t)d * H_DIM;
    unsafeAtomicAdd(&hd[lane],      w * hs[lane]);
    unsafeAtomicAdd(&hd[lane + 32], w * hs[lane + 32]);
}

// --------------------------- GEMM2 + bias + log_softmax (thread per node)
__global__ __launch_bounds__(256) void k_out(const float* __restrict__ h,
                                             const float* __restrict__ W2,
                                             const float* __restrict__ b2,
                                             float* __restrict__ out) {
    __shared__ float sW[H_DIM * C_OUT];
    __shared__ float sb[C_OUT];
    for (int i = threadIdx.x; i < H_DIM * C_OUT; i += 256) sW[i] = W2[i];
    if (threadIdx.x < C_OUT) sb[threadIdx.x] = b2[threadIdx.x];
    __syncthreads();

    int n = blockIdx.x * blockDim.x + threadIdx.x;
    if (n >= N_NODES) return;

    float hr[H_DIM];
    const float4* hp = (const float4*)(h + (size_t)n * H_DIM);
#pragma unroll
    for (int j = 0; j < H_DIM / 4; ++j) {
        float4 v = hp[j];
        hr[4 * j + 0] = v.x; hr[4 * j + 1] = v.y;
        hr[4 * j + 2] = v.z; hr[4 * j + 3] = v.w;
    }

    float logits[C_OUT];
    float mx = -1e30f;
#pragma unroll 4
    for (int c = 0; c < C_OUT; ++c) {
        float s = sb[c];
#pragma unroll
        for (int j = 0; j < H_DIM; ++j) s += hr[j] * sW[j * C_OUT + c];
        logits[c] = s;
        mx = fmaxf(mx, s);
    }
    float sum = 0.0f;
#pragma unroll 4
    for (int c = 0; c < C_OUT; ++c) sum += __expf(logits[c] - mx);
    float lse = mx + __logf(sum);
    float* op = out + (size_t)n * C_OUT;
#pragma unroll 4
    for (int c = 0; c < C_OUT; ++c) op[c] = logits[c] - lse;
}

// ---------------------------------------------------------------- launcher
extern "C" void kernel_launch(void* const* d_in, const int* in_sizes, int n_in,
                              void* d_out, int out_size, void* d_ws, size_t ws_size,
                              hipStream_t stream) {
    const float* x    = (const float*)d_in[0];
    const int*   ei   = (const int*)  d_in[1];
    const float* W1   = (const float*)d_in[2];
    const float* b1   = (const float*)d_in[3];
    const float* attl = (const float*)d_in[4];
    const float* attr = (const float*)d_in[5];
    const float* W2   = (const float*)d_in[6];
    const float* b2   = (const float*)d_in[7];
    float* out = (float*)d_out;

    float* ws  = (float*)d_ws;
    float* deg = ws;                       // N
    float* dis = deg + N_NODES;            // N
    float* al  = dis + N_NODES;            // N
    float* ar  = al  + N_NODES;            // N
    float* h0  = ar  + N_NODES;            // N*64
    float* hA  = h0  + (size_t)N_NODES * H_DIM;
    float* hB  = hA  + (size_t)N_NODES * H_DIM;

    // degree + norm
    k_zero<<<N_NODES / 256, 256, 0, stream>>>(deg, N_NODES);
    k_deg <<<E_EDGES / 256, 256, 0, stream>>>(ei, deg);
    k_dis <<<N_NODES / 256, 256, 0, stream>>>(deg, dis);

    // h0 = relu(x @ W1 + b1), WMMA f32
    k_gemm1<<<(N_NODES / 16) / 8, 256, 0, stream>>>(x, W1, b1, h0);

    // two FAConv layers (ping-pong h buffers)
    const float* hin = h0;
    float* hout = hA;
    for (int l = 0; l < 2; ++l) {
        k_att<<<(N_NODES * 32) / 256, 256, 0, stream>>>(
            hin, attl + l * H_DIM, attr + l * H_DIM, al, ar);
        k_scale_init<<<(N_NODES * H_DIM / 4) / 256, 256, 0, stream>>>(
            (const float4*)h0, (float4*)hout, N_NODES * H_DIM / 4);
        k_edge<<<(E_EDGES * 32) / 256, 256, 0, stream>>>(
            ei, dis, al, ar, hin, hout);
        hin = hout;
        hout = (l == 0) ? hB : hA;
    }

    // logits + log_softmax
    k_out<<<N_NODES / 256, 256, 0, stream>>>(hin, W2, b2, out);
}